// EncDecLSTM_29377576305346
// MI455X (gfx1250) — compile-verified
//
#include <hip/hip_runtime.h>

// EncDecLSTM fused persistent kernel for gfx1250 (MI455X).
// One wave32 per workgroup owns 16 batch rows through the entire enc/dec
// sequence (256 workgroups, 4 waves/SIMD via launch_bounds so VGPRs <= 256).
// fp32 WMMA (V_WMMA_F32_16X16X4_F32) for every GEMM. h/z state in LDS in a
// permuted A-operand layout (b128 reads, conflict-free), c state in registers
// in WMMA-C layout, GroupNorm via 16-lane shuffles. Weights + FiLM table are
// repacked once into d_ws so every B-operand / FiLM fetch is one coalesced
// global_load_b64. GEMMs are manually staged per 16-k group (load clause of
// 2 ds_b128 + 16 global_b64, then 16 chained WMMAs).

typedef __attribute__((ext_vector_type(2))) float v2f;
typedef __attribute__((ext_vector_type(4))) float v4f;
typedef __attribute__((ext_vector_type(8))) float v8f;

#define NB      4096
#define HIST_N  12
#define HOR_N   12
#define TSEQ    24      // HIST + HOR
#define DF      64
#define HDN     64
#define NTAIL   16
#define NCTX    12
#define LSTR    68      // padded LDS row stride (floats): 68%64==4 -> conflict-free

// ---- packed workspace layout (floats) ----
#define O_LAT   0
#define O_ENC   (O_LAT  + 64 * 64)          //  4096
#define O_DEC0  (O_ENC  + 256 * 256)        // 69632
#define O_DEC   (O_DEC0 + 64 * 256)         // 86016
#define O_TAIL  (O_DEC  + 128 * 256)        // 118784
#define O_EMB   (O_TAIL + 1024 * 64)        // 184320
#define WS_FLOATS (O_EMB + NCTX * 512)      // 190464
#define WS_BYTES  (WS_FLOATS * sizeof(float))

static __device__ __forceinline__ v8f wmma4(v2f a, v2f b, v8f c) {
  // D = A(16x4,f32) x B(4x16,f32) + C(16x16,f32)
  return __builtin_amdgcn_wmma_f32_16x16x4_f32(false, a, false, b, (short)0, c, false, false);
}

static __device__ __forceinline__ float sigm(float x)  { return 1.0f / (1.0f + __expf(-x)); }
static __device__ __forceinline__ float tanhx(float x) { return 2.0f / (1.0f + __expf(-2.0f * x)) - 1.0f; }

// Permuted LDS column layout for A-operand tiles (z / h):
// logical column c = t*16 + kk  ->  stored at t*16 + (kk&2)*4 + (kk>>2)*2 + (kk&1).
// A reader lane (r=n, hlf) then finds its 8 floats of each 16-k group contiguous
// at offset 8*hlf, enabling ds_load_b128.
static __device__ __forceinline__ int pcol(int t, int kk) {
  return t * 16 + ((kk & 2) << 2) + ((kk >> 2) << 1) + (kk & 1);
}

// ---------------- repack kernels (run once per launch, tiny) ----------------
// Row-pair interleave: dst[(k/2)*2N + n*2 + (k&1)] = src[k*N + n].
__global__ void EncDec_pack_w(float* __restrict__ dst, const float* __restrict__ src,
                              int ktot_n, int N) {
  int i = blockIdx.x * 256 + threadIdx.x;
  if (i >= ktot_n) return;
  int k = i / N, n = i - k * N;
  dst[(k >> 1) * (2 * N) + n * 2 + (k & 1)] = src[i];
}

// FiLM interleave: dst[c*512 + col*2 + {0,1}] = {emb[c*512+col], emb[c*512+256+col]}.
__global__ void EncDec_pack_emb(float* __restrict__ dst, const float* __restrict__ src) {
  int i = blockIdx.x * 256 + threadIdx.x;
  if (i >= NCTX * 256) return;
  int c = i >> 8, col = i & 255;
  dst[c * 512 + col * 2]     = src[c * 512 + col];
  dst[c * 512 + col * 2 + 1] = src[c * 512 + 256 + col];
}

// ---------------- building blocks ----------------
static __device__ __forceinline__ void bias_init(v8f acc[4], const float* __restrict__ bias,
                                                 int col0, int r) {
  #pragma unroll
  for (int t = 0; t < 4; ++t) {
    float bv = bias[col0 + t * 16 + r];
    #pragma unroll
    for (int v = 0; v < 8; ++v) acc[t][v] = bv;
  }
}

// A from LDS in permuted layout; B from W (P: row-pair packed -> b64/tile).
// Staged per 16-k group: all loads first, then 16 WMMAs.
template<int LDW, bool P>
static __device__ __forceinline__ void gemm64(v8f acc[4], const float* src,
                                              const float* __restrict__ W, int col0,
                                              int hlf, int r) {
  const float* ap = src + r * LSTR + 8 * hlf;
  #pragma unroll 1
  for (int kg = 0; kg < 64; kg += 16) {
    v4f q0 = *(const v4f*)(ap + kg);
    v4f q1 = *(const v4f*)(ap + kg + 4);
    v2f a[4];
    a[0].x = q0.x; a[0].y = q0.y;
    a[1].x = q0.z; a[1].y = q0.w;
    a[2].x = q1.x; a[2].y = q1.y;
    a[3].x = q1.z; a[3].y = q1.w;
    v2f b[4][4];
    #pragma unroll
    for (int j = 0; j < 4; ++j) {
      int kb = kg + j * 4;
      if (P) {
        const float* wp = W + ((kb >> 1) + hlf) * (2 * LDW) + (col0 + r) * 2;
        #pragma unroll
        for (int t = 0; t < 4; ++t) b[j][t] = *(const v2f*)(wp + t * 32);
      } else {
        const float* wp = W + (kb + 2 * hlf) * LDW + col0 + r;
        #pragma unroll
        for (int t = 0; t < 4; ++t) { b[j][t].x = wp[t * 16]; b[j][t].y = wp[LDW + t * 16]; }
      }
    }
    #pragma unroll
    for (int j = 0; j < 4; ++j) {
      #pragma unroll
      for (int t = 0; t < 4; ++t) acc[t] = wmma4(a[j], b[j][t], acc[t]);
    }
  }
}

// A directly from global x (row-major, row stride HIST*D); B = to_latent_W.
template<bool P>
static __device__ __forceinline__ void gemm_x(v8f acc[4], const float* __restrict__ xp,
                                              const float* __restrict__ W, int hlf, int r) {
  const float* ap = xp + (size_t)r * (HIST_N * DF) + 2 * hlf;
  #pragma unroll 1
  for (int kg = 0; kg < 64; kg += 16) {
    v2f a[4];
    #pragma unroll
    for (int j = 0; j < 4; ++j) a[j] = *(const v2f*)(ap + kg + j * 4);
    v2f b[4][4];
    #pragma unroll
    for (int j = 0; j < 4; ++j) {
      int kb = kg + j * 4;
      if (P) {
        const float* wp = W + ((kb >> 1) + hlf) * (2 * HDN) + r * 2;
        #pragma unroll
        for (int t = 0; t < 4; ++t) b[j][t] = *(const v2f*)(wp + t * 32);
      } else {
        const float* wp = W + (kb + 2 * hlf) * HDN + r;
        #pragma unroll
        for (int t = 0; t < 4; ++t) { b[j][t].x = wp[t * 16]; b[j][t].y = wp[HDN + t * 16]; }
      }
    }
    #pragma unroll
    for (int j = 0; j < 4; ++j) {
      #pragma unroll
      for (int t = 0; t < 4; ++t) acc[t] = wmma4(a[j], b[j][t], acc[t]);
    }
  }
}

// Per-row mean / rstd over a 16x64 tile held as 4 WMMA C tiles.
// Row m = v + 8*hlf lives in vgpr slot v, spread over 16 lanes (n) x 4 tiles.
static __device__ __forceinline__ void row_stats(const v8f a[4], float mu[8], float rs[8],
                                                 float eps) {
  #pragma unroll
  for (int v = 0; v < 8; ++v) {
    float s = a[0][v] + a[1][v] + a[2][v] + a[3][v];
    float q = a[0][v]*a[0][v] + a[1][v]*a[1][v] + a[2][v]*a[2][v] + a[3][v]*a[3][v];
    #pragma unroll
    for (int msk = 1; msk <= 8; msk <<= 1) {
      s += __shfl_xor(s, msk, 32);
      q += __shfl_xor(q, msk, 32);
    }
    float m_ = s * (1.0f / 64.0f);
    mu[v] = m_;
    rs[v] = rsqrtf(q * (1.0f / 64.0f) - m_ * m_ + eps);
  }
}

// GroupNorm(group of 64) + FiLM in place.
template<bool P>
static __device__ __forceinline__ void gn_film(v8f acc[4], const int* ctxid,
                                               const float* __restrict__ emb,
                                               int col0, int r, float eps) {
  float mu[8], rs[8];
  row_stats(acc, mu, rs, eps);
  #pragma unroll
  for (int v = 0; v < 8; ++v) {
    if (P) {
      const float* up = emb + (size_t)ctxid[v] * 512 + (col0 + r) * 2;
      #pragma unroll
      for (int t = 0; t < 4; ++t) {
        v2f ab = *(const v2f*)(up + t * 32);
        acc[t][v] = (acc[t][v] - mu[v]) * rs[v] * (1.0f + ab.x) + ab.y;
      }
    } else {
      const float* up = emb + (size_t)ctxid[v] * 512 + col0 + r;
      #pragma unroll
      for (int t = 0; t < 4; ++t) {
        float au = up[t * 16];
        float bu = up[256 + t * 16];
        acc[t][v] = (acc[t][v] - mu[v]) * rs[v] * (1.0f + au) + bu;
      }
    }
  }
}

// One LSTM cell. KD = 128 (zin = [z | h]) or 64 (zin = h only, dec0).
// ZMODE: 0 = don't touch z, 1 = z = h_new (dec0), 2 = z += h_new.
template<int KD, int ZMODE, bool P>
static __device__ __forceinline__ void lstm_cell(
    const float* zbuf, float* hbuf, float* zdst, v8f c[4],
    const float* __restrict__ W, const float* __restrict__ bias,
    const float* __restrict__ gamma, const float* __restrict__ beta,
    const float* __restrict__ emb, const int* ctxid,
    int hlf, int r) {
  const float eps = 1e-5f;
  v8f acc[4], tmp[4];

  // gate i (cols 0..63)
  bias_init(acc, bias, 0, r);
  if (KD == 128) { gemm64<256, P>(acc, zbuf, W, 0, hlf, r); gemm64<256, P>(acc, hbuf, W + 64 * 256, 0, hlf, r); }
  else           { gemm64<256, P>(acc, hbuf, W, 0, hlf, r); }
  gn_film<P>(acc, ctxid, emb, 0, r, eps);
  #pragma unroll
  for (int t = 0; t < 4; ++t) {
    #pragma unroll
    for (int v = 0; v < 8; ++v) tmp[t][v] = sigm(acc[t][v]);
  }

  // gate g (cols 192..255): tmp = sigmoid(i) * tanh(g)
  bias_init(acc, bias, 192, r);
  if (KD == 128) { gemm64<256, P>(acc, zbuf, W, 192, hlf, r); gemm64<256, P>(acc, hbuf, W + 64 * 256, 192, hlf, r); }
  else           { gemm64<256, P>(acc, hbuf, W, 192, hlf, r); }
  gn_film<P>(acc, ctxid, emb, 192, r, eps);
  #pragma unroll
  for (int t = 0; t < 4; ++t) {
    #pragma unroll
    for (int v = 0; v < 8; ++v) tmp[t][v] *= tanhx(acc[t][v]);
  }

  // gate f (cols 64..127): c = sigmoid(f)*c + tmp
  bias_init(acc, bias, 64, r);
  if (KD == 128) { gemm64<256, P>(acc, zbuf, W, 64, hlf, r); gemm64<256, P>(acc, hbuf, W + 64 * 256, 64, hlf, r); }
  else           { gemm64<256, P>(acc, hbuf, W, 64, hlf, r); }
  gn_film<P>(acc, ctxid, emb, 64, r, eps);
  #pragma unroll
  for (int t = 0; t < 4; ++t) {
    #pragma unroll
    for (int v = 0; v < 8; ++v) c[t][v] = sigm(acc[t][v]) * c[t][v] + tmp[t][v];
  }

  // gate o (cols 128..191): h = sigmoid(o) * tanh(GN1(c)*gamma + beta)
  bias_init(acc, bias, 128, r);
  if (KD == 128) { gemm64<256, P>(acc, zbuf, W, 128, hlf, r); gemm64<256, P>(acc, hbuf, W + 64 * 256, 128, hlf, r); }
  else           { gemm64<256, P>(acc, hbuf, W, 128, hlf, r); }
  gn_film<P>(acc, ctxid, emb, 128, r, eps);

  float mu2[8], rs2[8];
  row_stats(c, mu2, rs2, eps);
  #pragma unroll
  for (int t = 0; t < 4; ++t) {
    float g_ = gamma[t * 16 + r];
    float b_ = beta[t * 16 + r];
    int wc = pcol(t, r);
    #pragma unroll
    for (int v = 0; v < 8; ++v) {
      float gnc = (c[t][v] - mu2[v]) * rs2[v] * g_ + b_;
      float hv = sigm(acc[t][v]) * tanhx(gnc);
      int m = v + 8 * hlf;
      hbuf[m * LSTR + wc] = hv;                      // all hbuf reads finished above
      if (ZMODE == 1)      zdst[m * LSTR + wc] = hv;
      else if (ZMODE == 2) zdst[m * LSTR + wc] += hv;
    }
  }
}

template<bool P>
__global__ __launch_bounds__(32, 4)   // cap VGPRs at 256, 4 waves/SIMD
void EncDecLSTM_29377576305346_kernel(
    const float* __restrict__ x, const int* __restrict__ ctx,
    const float* __restrict__ latW, const float* __restrict__ latB,
    const float* __restrict__ encW, const float* __restrict__ encB,
    const float* __restrict__ encG, const float* __restrict__ encBe,
    const float* __restrict__ dec0W, const float* __restrict__ dec0B,
    const float* __restrict__ dec0G, const float* __restrict__ dec0Be,
    const float* __restrict__ decW, const float* __restrict__ decB,
    const float* __restrict__ decG, const float* __restrict__ decBe,
    const float* __restrict__ tailW, const float* __restrict__ tailB,
    const float* __restrict__ emb,
    float* __restrict__ out) {
  __shared__ float s_z[16 * LSTR];
  __shared__ float s_h[2][16 * LSTR];
  __shared__ int   s_ctx[16];

  const int lane = threadIdx.x & 31;
  const int hlf  = lane >> 4;
  const int r    = lane & 15;
  const int bbase = blockIdx.x * 16;

  float* zbuf = s_z;
  float* h0   = s_h[0];
  float* h1   = s_h[1];

  // zero h state
  for (int i = lane; i < 2 * 16 * LSTR; i += 32) (&s_h[0][0])[i] = 0.0f;
  v8f c0[4], c1[4];
  #pragma unroll
  for (int t = 0; t < 4; ++t) {
    #pragma unroll
    for (int v = 0; v < 8; ++v) { c0[t][v] = 0.0f; c1[t][v] = 0.0f; }
  }
  __syncthreads();

  int ctxid[8];

  // ---------------- encoder scan ----------------
  for (int t = 0; t < HIST_N; ++t) {
    if (lane < 16) s_ctx[lane] = ctx[(size_t)(bbase + lane) * TSEQ + t];
    __syncthreads();
    #pragma unroll
    for (int v = 0; v < 8; ++v) ctxid[v] = s_ctx[v + 8 * hlf];

    // z0 = x[:, t, :] @ to_latent_W + to_latent_b   (stored in permuted layout)
    {
      v8f acc[4];
      bias_init(acc, latB, 0, r);
      gemm_x<P>(acc, x + (size_t)bbase * HIST_N * DF + (size_t)t * DF, latW, hlf, r);
      #pragma unroll
      for (int tt = 0; tt < 4; ++tt) {
        int wc = pcol(tt, r);
        #pragma unroll
        for (int v = 0; v < 8; ++v)
          zbuf[(v + 8 * hlf) * LSTR + wc] = acc[tt][v];
      }
    }
    __syncthreads();

    lstm_cell<128, 2, P>(zbuf, h0, zbuf, c0, encW, encB, encG, encBe, emb, ctxid, hlf, r);
    __syncthreads();
    lstm_cell<128, 0, P>(zbuf, h1, zbuf, c1, encW + 128 * 256, encB + 256, encG + 64, encBe + 64,
                         emb, ctxid, hlf, r);
    __syncthreads();
  }

  // ---------------- decoder scan + tails ----------------
  for (int s = 0; s < HOR_N; ++s) {
    if (lane < 16) s_ctx[lane] = ctx[(size_t)(bbase + lane) * TSEQ + HIST_N + s];
    __syncthreads();
    #pragma unroll
    for (int v = 0; v < 8; ++v) ctxid[v] = s_ctx[v + 8 * hlf];

    lstm_cell<64, 1, P>(h0, h0, zbuf, c0, dec0W, dec0B, dec0G, dec0Be, emb, ctxid, hlf, r);
    __syncthreads();
    lstm_cell<128, 2, P>(zbuf, h1, zbuf, c1, decW, decB, decG, decBe, emb, ctxid, hlf, r);
    __syncthreads();

    // tails: out[b, n, s, :] = z[s, b, :] @ tail_W[n] + tail_b[n]
    for (int n = 0; n < NTAIL; ++n) {
      v8f acc[4];
      bias_init(acc, tailB + n * DF, 0, r);
      gemm64<64, P>(acc, zbuf, tailW + (size_t)n * HDN * DF, 0, hlf, r);
      #pragma unroll
      for (int tt = 0; tt < 4; ++tt) {
        #pragma unroll
        for (int v = 0; v < 8; ++v) {
          int m = v + 8 * hlf;
          out[((((size_t)(bbase + m)) * NTAIL + n) * HOR_N + s) * DF + tt * 16 + r] = acc[tt][v];
        }
      }
    }
    __syncthreads();
  }
}

extern "C" void kernel_launch(void* const* d_in, const int* in_sizes, int n_in,
                              void* d_out, int out_size, void* d_ws, size_t ws_size,
                              hipStream_t stream) {
  (void)in_sizes; (void)n_in; (void)out_size;
  const float* x     = (const float*)d_in[0];
  const int*   ctx   = (const int*)d_in[1];
  const float* latW  = (const float*)d_in[2];
  const float* latB  = (const float*)d_in[3];
  const float* encW  = (const float*)d_in[4];
  const float* encB  = (const float*)d_in[5];
  const float* encG  = (const float*)d_in[6];
  const float* encBe = (const float*)d_in[7];
  const float* dec0W = (const float*)d_in[8];
  const float* dec0B = (const float*)d_in[9];
  const float* dec0G = (const float*)d_in[10];
  const float* dec0Be= (const float*)d_in[11];
  const float* decW  = (const float*)d_in[12];
  const float* decB  = (const float*)d_in[13];
  const float* decG  = (const float*)d_in[14];
  const float* decBe = (const float*)d_in[15];
  const float* tailW = (const float*)d_in[16];
  const float* tailB = (const float*)d_in[17];
  const float* emb   = (const float*)d_in[18];
  float* out = (float*)d_out;

  dim3 block(32);            // one wave32 per workgroup
  dim3 grid(NB / 16);        // 256 workgroups, 16 batch rows each

  if (ws_size >= WS_BYTES) {
    float* ws = (float*)d_ws;
    float* latWp  = ws + O_LAT;
    float* encWp  = ws + O_ENC;
    float* dec0Wp = ws + O_DEC0;
    float* decWp  = ws + O_DEC;
    float* tailWp = ws + O_TAIL;
    float* embp   = ws + O_EMB;

    EncDec_pack_w<<<dim3((64 * 64 + 255) / 256), dim3(256), 0, stream>>>(latWp, latW, 64 * 64, 64);
    EncDec_pack_w<<<dim3((256 * 256 + 255) / 256), dim3(256), 0, stream>>>(encWp, encW, 256 * 256, 256);
    EncDec_pack_w<<<dim3((64 * 256 + 255) / 256), dim3(256), 0, stream>>>(dec0Wp, dec0W, 64 * 256, 256);
    EncDec_pack_w<<<dim3((128 * 256 + 255) / 256), dim3(256), 0, stream>>>(decWp, decW, 128 * 256, 256);
    EncDec_pack_w<<<dim3((1024 * 64 + 255) / 256), dim3(256), 0, stream>>>(tailWp, tailW, 1024 * 64, 64);
    EncDec_pack_emb<<<dim3((NCTX * 256 + 255) / 256), dim3(256), 0, stream>>>(embp, emb);

    EncDecLSTM_29377576305346_kernel<true><<<grid, block, 0, stream>>>(
        x, ctx, latWp, latB, encWp, encB, encG, encBe,
        dec0Wp, dec0B, dec0G, dec0Be, decWp, decB, decG, decBe,
        tailWp, tailB, embp, out);
  } else {
    EncDecLSTM_29377576305346_kernel<false><<<grid, block, 0, stream>>>(
        x, ctx, latW, latB, encW, encB, encG, encBe,
        dec0W, dec0B, dec0G, dec0Be, decW, decB, decG, decBe,
        tailW, tailB, emb, out);
  }
}